// GptqHinvCollector_64484638982673
// MI455X (gfx1250) — compile-verified
//
#include <hip/hip_runtime.h>

typedef __attribute__((ext_vector_type(2))) float v2f;
typedef __attribute__((ext_vector_type(8))) float v8f;

#define FDIM 1024
#define BS   128
#define NBLK 64

// ---------------- reduction: damp = 0.01 * mean(x^2) ----------------
__global__ void sumsq_partial_kernel(const float* __restrict__ x,
                                     float* __restrict__ partials, long n) {
  __shared__ float red[256];
  long stride = (long)gridDim.x * blockDim.x;
  long idx = (long)blockIdx.x * blockDim.x + threadIdx.x;
  float s = 0.f;
  for (long i = idx; i < n; i += stride) { float v = x[i]; s += v * v; }
  red[threadIdx.x] = s;
  __syncthreads();
  for (int off = 128; off > 0; off >>= 1) {
    if ((int)threadIdx.x < off) red[threadIdx.x] += red[threadIdx.x + off];
    __syncthreads();
  }
  if (threadIdx.x == 0) partials[blockIdx.x] = red[0];
}

__global__ void finalize_damp_kernel(const float* __restrict__ partials,
                                     float* __restrict__ dampInv, long n, int nparts) {
  __shared__ float red[256];
  float s = 0.f;
  for (int i = threadIdx.x; i < nparts; i += 256) s += partials[i];
  red[threadIdx.x] = s;
  __syncthreads();
  for (int off = 128; off > 0; off >>= 1) {
    if ((int)threadIdx.x < off) red[threadIdx.x] += red[threadIdx.x + off];
    __syncthreads();
  }
  if (threadIdx.x == 0) {
    float mean = red[0] / (float)n;
    dampInv[0] = 1.0f / (0.01f * mean);
  }
}

__global__ void init_hinv_kernel(float* __restrict__ hinv,
                                 const float* __restrict__ dampInv) {
  long idx = (long)blockIdx.x * blockDim.x + threadIdx.x;
  if (idx < (long)FDIM * FDIM) {
    long i = idx / FDIM, j = idx % FDIM;
    hinv[idx] = (i == j) ? dampInv[0] : 0.0f;
  }
}

__global__ void perm_kernel(int* __restrict__ perm) {
  int i = blockIdx.x * blockDim.x + threadIdx.x;
  if (i < FDIM) perm[i] = i;
}

// ---------------- fp32 WMMA GEMM, unit K strides, 32x32 tile per wave -------
// D[m, n] (row-major, ldc) = prodScale * sum_k A[m*sAm + k] * B[n*sBn + k]
//                            + (Cin ? Cin[m*ldc+n] : 0) + (addEye && m==n)
// NB: the B operand is indexed (n, k) — i.e. this computes A * B^T where both
// operands are k-contiguous, so fragment loads are aligned 8-byte v2f loads.
// Each wave32 computes a 32x32 tile via 4 accumulators; 4 waves per block.
__global__ void __launch_bounds__(128)
wmma_gemm_f32_kernel(const float* __restrict__ A, long sAm,
                     const float* __restrict__ B, long sBn,
                     const float* __restrict__ Cin, float* __restrict__ D,
                     long ldc, int tilesN, int K, float prodScale, int addEye) {
  const int lane  = threadIdx.x & 31;
  const int wave  = threadIdx.x >> 5;
  const int tile  = blockIdx.x * 4 + wave;   // 32x32 tiles
  const int tm    = tile / tilesN;
  const int tn    = tile % tilesN;
  const int mrow  = lane & 15;               // row within A-frag / col within B-frag
  const int khalf = (lane >> 4) << 1;        // lanes 16..31 hold k+2, k+3

  const long m0 = (long)tm * 32;
  const long n0 = (long)tn * 32;

  v8f acc00 = {0.f,0.f,0.f,0.f,0.f,0.f,0.f,0.f};
  v8f acc01 = acc00, acc10 = acc00, acc11 = acc00;

  const float* A0 = A + (m0 + mrow) * sAm + khalf;   // 8-byte aligned (khalf even)
  const float* A1 = A0 + 16 * sAm;
  const float* B0 = B + (n0 + mrow) * sBn + khalf;
  const float* B1 = B0 + 16 * sBn;

  for (int k = 0; k < K; k += 4) {
    v2f a0 = *(const v2f*)(A0 + k);
    v2f a1 = *(const v2f*)(A1 + k);
    v2f b0 = *(const v2f*)(B0 + k);
    v2f b1 = *(const v2f*)(B1 + k);
    acc00 = __builtin_amdgcn_wmma_f32_16x16x4_f32(false, a0, false, b0, (short)0, acc00, false, false);
    acc01 = __builtin_amdgcn_wmma_f32_16x16x4_f32(false, a0, false, b1, (short)0, acc01, false, false);
    acc10 = __builtin_amdgcn_wmma_f32_16x16x4_f32(false, a1, false, b0, (short)0, acc10, false, false);
    acc11 = __builtin_amdgcn_wmma_f32_16x16x4_f32(false, a1, false, b1, (short)0, acc11, false, false);
  }

  // C/D layout: VGPR r -> M = r (+8 for lanes 16..31), N = lane & 15
  const long cn = (long)(lane & 15);
  const long cm = (long)((lane >> 4) << 3);
  auto store_tile = [&](const v8f& acc, long mbase, long nbase) {
    long ccol = nbase + cn;
    for (int r = 0; r < 8; ++r) {
      long crow = mbase + cm + r;
      float v = prodScale * acc[r];
      if (Cin) v += Cin[crow * ldc + ccol];
      if (addEye && crow == ccol) v += 1.0f;
      D[crow * ldc + ccol] = v;
    }
  };
  store_tile(acc00, m0,      n0);
  store_tile(acc01, m0,      n0 + 16);
  store_tile(acc10, m0 + 16, n0);
  store_tile(acc11, m0 + 16, n0 + 16);
}

// ---------------- 128x128 in-place Cholesky (lower), one workgroup ----------
__global__ void __launch_bounds__(BS) cholesky128_kernel(float* __restrict__ S) {
  const int tid = threadIdx.x;  // one thread per row
  for (int j = 0; j < BS; ++j) {
    if (tid == j) S[j * BS + j] = sqrtf(S[j * BS + j]);
    __syncthreads();
    float dinv = 1.0f / S[j * BS + j];
    if (tid > j) S[tid * BS + j] *= dinv;
    __syncthreads();
    if (tid > j) {
      float ljk = S[tid * BS + j];
      for (int c = j + 1; c <= tid; ++c)
        S[tid * BS + c] -= ljk * S[c * BS + j];
    }
    __syncthreads();
  }
}

// ---------------- forward solve: Ut(f,i) = (L^-1 * HB^T)(i,f) ---------------
// HBt is (BS x F) row-major with HBt[i][f] = HB[f][i]; RHS reads are
// lane-coalesced. One column f per thread, running solution in 64 KB LDS
// (conflict-free: consecutive lanes -> consecutive banks), L reads uniform.
// Output Ut is (F x BS) row-major, k-contiguous for the SYRK consumer.
__global__ void __launch_bounds__(BS)
trsm_fwd_kernel(const float* __restrict__ L, const float* __restrict__ HBt,
                float* __restrict__ Ut) {
  __shared__ float u[BS * BS];
  const int tid = threadIdx.x;
  const long f  = (long)blockIdx.x * BS + tid;  // column 0..1023
  for (int i = 0; i < BS; ++i) {
    float s = HBt[(long)i * FDIM + f];
    for (int p = 0; p < i; ++p)
      s -= L[i * BS + p] * u[p * BS + tid];
    s /= L[i * BS + i];
    u[i * BS + tid] = s;
    Ut[f * BS + i] = s;
  }
}

extern "C" void kernel_launch(void* const* d_in, const int* in_sizes, int n_in,
                              void* d_out, int out_size, void* d_ws, size_t ws_size,
                              hipStream_t stream) {
  (void)n_in; (void)out_size; (void)ws_size;
  const float* x = (const float*)d_in[0];
  float* hinv = (float*)d_out;                              // 1024*1024 f32
  int*   perm = (int*)((float*)d_out + (long)FDIM * FDIM);  // 1024 i32

  char*  ws       = (char*)d_ws;
  float* dampInv  = (float*)(ws);                                   // 1 f32
  float* partials = (float*)(ws + 256);                             // 1024 f32
  float* HBt      = (float*)(ws + 8192);                            // BS*F
  float* S        = (float*)(ws + 8192 + (long)BS * FDIM * 4);      // BS*BS
  float* Ut       = (float*)(ws + 8192 + (long)BS * FDIM * 4 + (long)BS * BS * 4); // F*BS

  const long n = (long)in_sizes[0];  // 4*2048*1024

  sumsq_partial_kernel<<<1024, 256, 0, stream>>>(x, partials, n);
  finalize_damp_kernel<<<1, 256, 0, stream>>>(partials, dampInv, n, 1024);
  init_hinv_kernel<<<(FDIM * FDIM + 255) / 256, 256, 0, stream>>>(hinv, dampInv);
  perm_kernel<<<(FDIM + 255) / 256, 256, 0, stream>>>(perm);

  for (int blk = 0; blk < NBLK; ++blk) {
    const float* xblk = x + (long)blk * BS * FDIM;  // B: (BS x F) row-major

    // HBt = B @ hinv  (hinv symmetric => B-operand read k-contiguous)
    // M=BS, N=F, K=F:  A(m,k)=xblk[m*F+k], B(n,k)=hinv[n*F+k]
    wmma_gemm_f32_kernel<<<(BS / 32) * (FDIM / 32) / 4, 128, 0, stream>>>(
        xblk, (long)FDIM,
        hinv, (long)FDIM,
        nullptr, HBt, (long)FDIM,
        FDIM / 32, FDIM, 1.0f, 0);

    // S = I + B @ HBt^T : M=BS, N=BS, K=F
    //   A(m,k)=xblk[m*F+k], B(n,k)=HBt[n*F+k]
    wmma_gemm_f32_kernel<<<(BS / 32) * (BS / 32) / 4, 128, 0, stream>>>(
        xblk, (long)FDIM,
        HBt, (long)FDIM,
        nullptr, S, (long)BS,
        BS / 32, FDIM, 1.0f, 1);

    cholesky128_kernel<<<1, BS, 0, stream>>>(S);

    trsm_fwd_kernel<<<FDIM / BS, BS, 0, stream>>>(S, HBt, Ut);

    // hinv -= U^T @ U = Ut @ Ut^T : M=F, N=F, K=BS
    //   A(m,k)=Ut[m*BS+k], B(n,k)=Ut[n*BS+k]
    wmma_gemm_f32_kernel<<<(FDIM / 32) * (FDIM / 32) / 4, 128, 0, stream>>>(
        Ut, (long)BS,
        Ut, (long)BS,
        hinv, hinv, (long)FDIM,
        FDIM / 32, BS, -1.0f, 0);
  }
}